// GLFB_1013612282254
// MI455X (gfx1250) — compile-verified
//
#include <hip/hip_runtime.h>

// ---------------------------------------------------------------------------
// GLFB block for MI455X (gfx1250, wave32, WMMA).
// Uses V_WMMA_F32_16X16X4_F32 for all channel GEMMs (fp32-exact path).
//
// Tensor layout: x[b][c][n], n = f*S+s, NSP = 65536, per-batch stride CN.
// Each wave owns one 16-column spatial tile and all 64 input channels (K).
// B-panel (K=64 x N=16) kept in 16 v2f register fragments.
// Assumed 16x16x4 fragment layouts (mirrors ISA 16x4 A table):
//   A (16x4): v0 = {K=0 | K=2}, v1 = {K=1 | K=3} across lane halves, M=lane&15
//   B (4x16): v0 = {row K=0 | row K=2}, v1 = {K=1 | K=3}, N=lane&15
//   D (16x16): vgpr r -> M = r + (lane>=16 ? 8 : 0), N = lane&15
// ---------------------------------------------------------------------------

typedef __attribute__((ext_vector_type(2))) float v2f;
typedef __attribute__((ext_vector_type(8))) float v8f;

#define CCH   64
#define CCH2  128
#define NSP   65536
#define BATCH 4
#define CN    (CCH * NSP)        // 4,194,304 elements per batch
#define LNEPS 1e-5f
#define WPAD  66                 // LDS row stride for weight tiles (bank-safe)
#define GPAD  68                 // LDS col stride for GLU scratch (bank-safe)

// ws layout (floats):
//  [0..3]    ln1 sum per batch        [4..7]    ln1 sumsq
//  [8..11]   ln2 sum per batch        [12..15]  ln2 sumsq
//  [16..271] h spatial-pool psum [4][64]
//  [272..527] SE attention scale s [4][64]

__device__ __forceinline__ float sigmoidf_(float v) {
    return 1.0f / (1.0f + __expf(-v));
}

__global__ void k_zero(float* ws) {
    int t = blockIdx.x * 256 + threadIdx.x;
    if (t < 528) ws[t] = 0.0f;
}

// ---- K1: per-batch sum / sumsq of x (for LayerNorm 1) ----------------------
__global__ void k_stats_x(const float* __restrict__ x, float* __restrict__ ws) {
    // grid = BATCH*512 blocks of 256; each thread sums 8 float4 (32 floats)
    int blk   = blockIdx.x;
    int batch = blk >> 9;
    int lb    = blk & 511;
    const float4* xb = (const float4*)(x + (size_t)batch * CN);
    int base = lb * (256 * 8) + threadIdx.x;
    float s = 0.0f, q = 0.0f;
#pragma unroll
    for (int i = 0; i < 8; ++i) {
        float4 v = xb[base + i * 256];
        s += v.x + v.y + v.z + v.w;
        q += v.x * v.x + v.y * v.y + v.z * v.z + v.w * v.w;
    }
#pragma unroll
    for (int m = 16; m >= 1; m >>= 1) {
        s += __shfl_xor(s, m, 32);
        q += __shfl_xor(q, m, 32);
    }
    __shared__ float ls[8], lq[8];
    int wid = threadIdx.x >> 5;
    if ((threadIdx.x & 31) == 0) { ls[wid] = s; lq[wid] = q; }
    __syncthreads();
    if (threadIdx.x == 0) {
        float S = 0.0f, Q = 0.0f;
#pragma unroll
        for (int i = 0; i < 8; ++i) { S += ls[i]; Q += lq[i]; }
        atomicAdd(&ws[batch], S);
        atomicAdd(&ws[4 + batch], Q);
    }
}

// ---- K2: LN1 -> GEMM w1 (64->128) -> depthwise affine -> GLU -> h ----------
//      also accumulates spatial pool of h for channel attention.
__global__ void __launch_bounds__(256)
k_branch1a(const float* __restrict__ x, const float* __restrict__ lnw,
           const float* __restrict__ lnb, const float* __restrict__ w1,
           const float* __restrict__ b1, const float* __restrict__ dww,
           const float* __restrict__ dwb, float* __restrict__ hout,
           float* __restrict__ ws) {
    __shared__ float Wl[CCH2 * WPAD];
    for (int i = threadIdx.x; i < CCH2 * CCH; i += 256)
        Wl[(i >> 6) * WPAD + (i & 63)] = w1[i];
    __syncthreads();

    int lane  = threadIdx.x & 31;
    int wave  = threadIdx.x >> 5;
    int tile  = blockIdx.x * 8 + wave;
    int batch = tile >> 12;                 // 4096 tiles per batch
    int col   = ((tile & 4095) << 4) + (lane & 15);
    int kg    = lane >> 4;                  // 0 or 1: which K pair
    size_t bbase = (size_t)batch * CN;

    float invM = 1.0f / (float)CN;
    float mean = ws[batch] * invM;
    float var  = ws[4 + batch] * invM - mean * mean;
    float rstd = rsqrtf(var + LNEPS);

    // B-panel: normalized input, K=64 x N=16 per wave
    v2f Bv[16];
#pragma unroll
    for (int kk = 0; kk < 16; ++kk) {
        int k0 = kk * 4 + kg * 2;
        size_t o0 = (size_t)k0 * NSP + col;
        float v0 = (x[bbase + o0] - mean) * rstd * lnw[o0] + lnb[o0];
        float v1 = (x[bbase + o0 + NSP] - mean) * rstd * lnw[o0 + NSP] + lnb[o0 + NSP];
        Bv[kk] = v2f{v0, v1};
    }

    int moff = kg ? 8 : 0;
#pragma unroll
    for (int m = 0; m < 4; ++m) {
        v8f accA = {};
        v8f accG = {};
#pragma unroll
        for (int kk = 0; kk < 16; ++kk) {
            int kbase = kk * 4 + kg * 2;
            v2f af = *(const v2f*)&Wl[(m * 16 + (lane & 15)) * WPAD + kbase];
            v2f gf = *(const v2f*)&Wl[((m + 4) * 16 + 64 + (lane & 15) - 64) * WPAD + kbase + (CCH2 / 2) * 0];
            gf = *(const v2f*)&Wl[((m + 4) * 16 + (lane & 15)) * WPAD + kbase];
            accA = __builtin_amdgcn_wmma_f32_16x16x4_f32(
                false, af, false, Bv[kk], (short)0, accA, false, false);
            accG = __builtin_amdgcn_wmma_f32_16x16x4_f32(
                false, gf, false, Bv[kk], (short)0, accG, false, false);
        }
#pragma unroll
        for (int r = 0; r < 8; ++r) {
            int rowA = m * 16 + r + moff;         // 0..63  (a-half)
            int rowG = 64 + rowA;                 // 64..127 (gate-half)
            float va = accA[r] + b1[rowA];
            va = va * dww[rowA] + dwb[rowA];
            float vg = accG[r] + b1[rowG];
            vg = vg * dww[rowG] + dwb[rowG];
            float h = va * sigmoidf_(vg);
            hout[bbase + (size_t)rowA * NSP + col] = h;
            // reduce across the 16-lane half (same rowA within a half)
            float ps = h;
            ps += __shfl_xor(ps, 1, 32);
            ps += __shfl_xor(ps, 2, 32);
            ps += __shfl_xor(ps, 4, 32);
            ps += __shfl_xor(ps, 8, 32);
            if ((lane & 15) == 0)
                atomicAdd(&ws[16 + batch * 64 + rowA], ps);
        }
    }
}

// ---- K3: tiny SE MLP: s = sigmoid(relu(p W1a^T + b) W2a^T + b) -------------
__global__ void k_attn(const float* __restrict__ aw1, const float* __restrict__ ab1,
                       const float* __restrict__ aw2, const float* __restrict__ ab2,
                       float* __restrict__ ws) {
    __shared__ float p[256], r[256];
    int t = threadIdx.x;
    int b = t >> 6, o = t & 63;
    p[t] = ws[16 + t] * (1.0f / (float)NSP);
    __syncthreads();
    float acc = ab1[o];
#pragma unroll 8
    for (int c = 0; c < 64; ++c) acc += p[b * 64 + c] * aw1[o * 64 + c];
    r[t] = fmaxf(acc, 0.0f);
    __syncthreads();
    float acc2 = ab2[o];
#pragma unroll 8
    for (int c = 0; c < 64; ++c) acc2 += r[b * 64 + c] * aw2[o * 64 + c];
    ws[272 + t] = sigmoidf_(acc2);
}

// ---- K4: y = (h * s) @ w2^T + b2 + x, in place over h; + LN2 stats ---------
__global__ void __launch_bounds__(256)
k_branch1b(const float* __restrict__ x, const float* __restrict__ w2,
           const float* __restrict__ b2, float* __restrict__ hy,
           float* __restrict__ ws) {
    __shared__ float Wl[CCH * WPAD];
    __shared__ float sS[64];
    int lane  = threadIdx.x & 31;
    int wave  = threadIdx.x >> 5;
    int tile  = blockIdx.x * 8 + wave;
    int batch = tile >> 12;                 // same batch for whole block
    for (int i = threadIdx.x; i < CCH * CCH; i += 256)
        Wl[(i >> 6) * WPAD + (i & 63)] = w2[i];
    if (threadIdx.x < 64) sS[threadIdx.x] = ws[272 + batch * 64 + threadIdx.x];
    __syncthreads();

    int col = ((tile & 4095) << 4) + (lane & 15);
    int kg  = lane >> 4;
    size_t bbase = (size_t)batch * CN;

    v2f Bv[16];
#pragma unroll
    for (int kk = 0; kk < 16; ++kk) {
        int k0 = kk * 4 + kg * 2;
        size_t o0 = bbase + (size_t)k0 * NSP + col;
        Bv[kk] = v2f{hy[o0] * sS[k0], hy[o0 + NSP] * sS[k0 + 1]};
    }

    int moff = kg ? 8 : 0;
    float lsum = 0.0f, lsq = 0.0f;
#pragma unroll
    for (int m = 0; m < 4; ++m) {
        v8f acc = {};
#pragma unroll
        for (int kk = 0; kk < 16; ++kk) {
            int kbase = kk * 4 + kg * 2;
            v2f af = *(const v2f*)&Wl[(m * 16 + (lane & 15)) * WPAD + kbase];
            acc = __builtin_amdgcn_wmma_f32_16x16x4_f32(
                false, af, false, Bv[kk], (short)0, acc, false, false);
        }
#pragma unroll
        for (int r = 0; r < 8; ++r) {
            int row = m * 16 + r + moff;
            size_t off = bbase + (size_t)row * NSP + col;
            float v = acc[r] + b2[row] + x[off];
            hy[off] = v;                       // y overwrites h in place
            lsum += v;
            lsq  += v * v;
        }
    }
#pragma unroll
    for (int msk = 16; msk >= 1; msk >>= 1) {
        lsum += __shfl_xor(lsum, msk, 32);
        lsq  += __shfl_xor(lsq,  msk, 32);
    }
    if (lane == 0) {
        atomicAdd(&ws[8 + batch],  lsum);
        atomicAdd(&ws[12 + batch], lsq);
    }
}

// ---- K6: LN2 -> GEMM w3 (64->128) -> GLU -> GEMM w4 (64->64) -> + y --------
//      (in place over y; GLU tile re-shaped via wave-private padded LDS)
__global__ void __launch_bounds__(256)
k_branch2(const float* __restrict__ lnw, const float* __restrict__ lnb,
          const float* __restrict__ w3, const float* __restrict__ b3,
          const float* __restrict__ w4, const float* __restrict__ b4,
          float* __restrict__ y, float* __restrict__ ws) {
    __shared__ float W3l[CCH2 * WPAD];
    __shared__ float W4l[CCH * WPAD];
    __shared__ float Gl[8 * 16 * GPAD];     // per-wave GLU scratch, col-major
    for (int i = threadIdx.x; i < CCH2 * CCH; i += 256)
        W3l[(i >> 6) * WPAD + (i & 63)] = w3[i];
    for (int i = threadIdx.x; i < CCH * CCH; i += 256)
        W4l[(i >> 6) * WPAD + (i & 63)] = w4[i];
    __syncthreads();

    int lane  = threadIdx.x & 31;
    int wave  = threadIdx.x >> 5;
    int tile  = blockIdx.x * 8 + wave;
    int batch = tile >> 12;
    int col   = ((tile & 4095) << 4) + (lane & 15);
    int kg    = lane >> 4;
    size_t bbase = (size_t)batch * CN;

    float invM = 1.0f / (float)CN;
    float mean = ws[8 + batch] * invM;
    float var  = ws[12 + batch] * invM - mean * mean;
    float rstd = rsqrtf(var + LNEPS);

    v2f Bv[16];
#pragma unroll
    for (int kk = 0; kk < 16; ++kk) {
        int k0 = kk * 4 + kg * 2;
        size_t o0 = (size_t)k0 * NSP + col;
        float v0 = (y[bbase + o0] - mean) * rstd * lnw[o0] + lnb[o0];
        float v1 = (y[bbase + o0 + NSP] - mean) * rstd * lnw[o0 + NSP] + lnb[o0 + NSP];
        Bv[kk] = v2f{v0, v1};
    }

    float* gw = &Gl[wave * 16 * GPAD];      // wave-private: no barrier needed
    int moff = kg ? 8 : 0;
#pragma unroll
    for (int m = 0; m < 4; ++m) {
        v8f accA = {};
        v8f accG = {};
#pragma unroll
        for (int kk = 0; kk < 16; ++kk) {
            int kbase = kk * 4 + kg * 2;
            v2f af = *(const v2f*)&W3l[(m * 16 + (lane & 15)) * WPAD + kbase];
            v2f gf = *(const v2f*)&W3l[((m + 4) * 16 + (lane & 15)) * WPAD + kbase];
            accA = __builtin_amdgcn_wmma_f32_16x16x4_f32(
                false, af, false, Bv[kk], (short)0, accA, false, false);
            accG = __builtin_amdgcn_wmma_f32_16x16x4_f32(
                false, gf, false, Bv[kk], (short)0, accG, false, false);
        }
#pragma unroll
        for (int r = 0; r < 8; ++r) {
            int rowA = m * 16 + r + moff;
            float va = accA[r] + b3[rowA];
            float vg = accG[r] + b3[64 + rowA];
            gw[(lane & 15) * GPAD + rowA] = va * sigmoidf_(vg);
        }
    }

    // Reload GLU tile as B-fragments (same wave: DS ops in order, waits auto)
    v2f B2[16];
#pragma unroll
    for (int kk = 0; kk < 16; ++kk) {
        int k0 = kk * 4 + kg * 2;
        B2[kk] = *(const v2f*)&gw[(lane & 15) * GPAD + k0];
    }
#pragma unroll
    for (int m = 0; m < 4; ++m) {
        v8f acc = {};
#pragma unroll
        for (int kk = 0; kk < 16; ++kk) {
            int kbase = kk * 4 + kg * 2;
            v2f af = *(const v2f*)&W4l[(m * 16 + (lane & 15)) * WPAD + kbase];
            acc = __builtin_amdgcn_wmma_f32_16x16x4_f32(
                false, af, false, B2[kk], (short)0, acc, false, false);
        }
#pragma unroll
        for (int r = 0; r < 8; ++r) {
            int row = m * 16 + r + moff;
            size_t off = bbase + (size_t)row * NSP + col;
            y[off] = acc[r] + b4[row] + y[off];   // out = h2 + y, in place
        }
    }
}

// ---------------------------------------------------------------------------
extern "C" void kernel_launch(void* const* d_in, const int* in_sizes, int n_in,
                              void* d_out, int out_size, void* d_ws, size_t ws_size,
                              hipStream_t stream) {
    const float* x    = (const float*)d_in[0];
    const float* ln1w = (const float*)d_in[1];
    const float* ln1b = (const float*)d_in[2];
    const float* ln2w = (const float*)d_in[3];
    const float* ln2b = (const float*)d_in[4];
    const float* w1   = (const float*)d_in[5];
    const float* b1   = (const float*)d_in[6];
    const float* dww  = (const float*)d_in[7];
    const float* dwb  = (const float*)d_in[8];
    const float* aw1  = (const float*)d_in[9];
    const float* ab1  = (const float*)d_in[10];
    const float* aw2  = (const float*)d_in[11];
    const float* ab2  = (const float*)d_in[12];
    const float* w2   = (const float*)d_in[13];
    const float* b2   = (const float*)d_in[14];
    const float* w3   = (const float*)d_in[15];
    const float* b3   = (const float*)d_in[16];
    const float* w4   = (const float*)d_in[17];
    const float* b4   = (const float*)d_in[18];
    float* out = (float*)d_out;                 // used in place: h -> y -> out
    float* ws  = (float*)d_ws;

    const int TILES = BATCH * (NSP / 16);       // 16384 wave tiles
    k_zero<<<3, 256, 0, stream>>>(ws);
    k_stats_x<<<BATCH * 512, 256, 0, stream>>>(x, ws);
    k_branch1a<<<TILES / 8, 256, 0, stream>>>(x, ln1w, ln1b, w1, b1, dww, dwb,
                                              out, ws);
    k_attn<<<1, 256, 0, stream>>>(aw1, ab1, aw2, ab2, ws);
    k_branch1b<<<TILES / 8, 256, 0, stream>>>(x, w2, b2, out, ws);
    k_branch2<<<TILES / 8, 256, 0, stream>>>(ln2w, ln2b, w3, b3, w4, b4, out, ws);
}